// EdgeTypeGAT_16260746183178
// MI455X (gfx1250) — compile-verified
//
#include <hip/hip_runtime.h>

// GAT layer for MI455X (gfx1250), wave32.
// Pipeline: WMMA f32 projection GEMM -> attn logits -> segment softmax via
// ordered-uint atomicMax + atomicAdd (L2-resident tables) -> scatter aggregate.

typedef float v2f __attribute__((ext_vector_type(2)));
typedef float v8f __attribute__((ext_vector_type(8)));

#define SA_STRIDE 68   // 16x64 A tile padded: conflict-free b64 reads
#define SW_STRIDE 68   // W^T tile padded likewise

// ---------------------------------------------------------------------------
// Pass A: feat[N,64] = nf[N,64] @ W[64,64] using V_WMMA_F32_16X16X4_F32.
// 256 threads = 8 waves; each wave owns a 16-row tile (block covers 128 rows).
// A frag (16x4 f32): lanes 0-15: M=lane, {K0,K1}; lanes 16-31: M=lane-16, {K2,K3}
// B frag (4x16 f32): lanes 0-15: N=lane, {K0,K1}; lanes 16-31: N=lane-16, {K2,K3}
// C/D (16x16 f32):   vgpr v, lanes 0-15: M=v, N=lane; lanes 16-31: M=v+8
// W is staged TRANSPOSED in LDS so each B fragment is one contiguous
// 8-byte-aligned pair -> single ds_load_b64 into an even VGPR pair.
// ---------------------------------------------------------------------------
__global__ __launch_bounds__(256) void gat_proj_wmma(
    const float* __restrict__ nf, const float* __restrict__ W,
    float* __restrict__ feat, int N)
{
    __shared__ float sWT[64 * SW_STRIDE];          // sWT[n*68 + k] = W[k][n]
    __shared__ float sA[8][16 * SA_STRIDE];

    const int tid  = threadIdx.x;
    const int wave = tid >> 5;
    const int lane = tid & 31;

    // Stage W transposed (coalesced global reads; one-time LDS writes).
    for (int i = tid; i < 64 * 64; i += 256) {
        const int k = i >> 6, n = i & 63;
        sWT[n * SW_STRIDE + k] = W[i];
    }

    // Stage this wave's 16x64 A tile (coalesced 128B global loads).
    const int rowbase = blockIdx.x * 128 + wave * 16;
    for (int r = 0; r < 16; ++r) {
        int row = rowbase + r;
        row = row < N ? row : (N - 1);             // clamp; writes are guarded
        sA[wave][r * SA_STRIDE + lane]      = nf[(size_t)row * 64 + lane];
        sA[wave][r * SA_STRIDE + lane + 32] = nf[(size_t)row * 64 + lane + 32];
    }
    __syncthreads();

    const int half = lane >> 4;      // 0: {K0,K1}, 1: {K2,K3}
    const int m    = lane & 15;      // row (A) / col (B) within 16-wide tile

    v8f acc0 = {}, acc1 = {}, acc2 = {}, acc3 = {};

    #pragma unroll
    for (int k0 = 0; k0 < 64; k0 += 4) {
        const int ka = k0 + half * 2;

        const v2f a  = *(const v2f*)&sA[wave][m * SA_STRIDE + ka];
        const v2f b0 = *(const v2f*)&sWT[(m +  0) * SW_STRIDE + ka];
        const v2f b1 = *(const v2f*)&sWT[(m + 16) * SW_STRIDE + ka];
        const v2f b2 = *(const v2f*)&sWT[(m + 32) * SW_STRIDE + ka];
        const v2f b3 = *(const v2f*)&sWT[(m + 48) * SW_STRIDE + ka];

        acc0 = __builtin_amdgcn_wmma_f32_16x16x4_f32(false, a, false, b0, (short)0, acc0, false, false);
        acc1 = __builtin_amdgcn_wmma_f32_16x16x4_f32(false, a, false, b1, (short)0, acc1, false, false);
        acc2 = __builtin_amdgcn_wmma_f32_16x16x4_f32(false, a, false, b2, (short)0, acc2, false, false);
        acc3 = __builtin_amdgcn_wmma_f32_16x16x4_f32(false, a, false, b3, (short)0, acc3, false, false);
    }

    #pragma unroll
    for (int v = 0; v < 8; ++v) {
        const int row = rowbase + v + half * 8;
        if (row < N) {
            float* fr = feat + (size_t)row * 64;
            fr[ 0 + m] = acc0[v];
            fr[16 + m] = acc1[v];
            fr[32 + m] = acc2[v];
            fr[48 + m] = acc3[v];
        }
    }
}

// ---------------------------------------------------------------------------
// Pass B: per-(node,head) attention logits el/er.
// ---------------------------------------------------------------------------
__global__ void gat_elr(const float* __restrict__ feat,
                        const float* __restrict__ al, const float* __restrict__ ar,
                        float* __restrict__ el, float* __restrict__ er, int N)
{
    int t = blockIdx.x * blockDim.x + threadIdx.x;
    if (t >= N * 4) return;
    const int n = t >> 2, h = t & 3;
    const float* f = feat + (size_t)n * 64 + h * 16;
    float sl = 0.f, sr = 0.f;
    #pragma unroll
    for (int i = 0; i < 16; ++i) {
        const float v = f[i];
        sl += v * al[h * 16 + i];
        sr += v * ar[h * 16 + i];
    }
    el[t] = sl;
    er[t] = sr;
}

// ---------------------------------------------------------------------------
// Init: out <- bias broadcast; denom <- 0; emax (ordered-uint) <- 0 sentinel.
// ---------------------------------------------------------------------------
__global__ void gat_init(float* __restrict__ out, const float* __restrict__ bias,
                         float* __restrict__ denom, unsigned* __restrict__ emax_u, int N)
{
    int t = blockIdx.x * blockDim.x + threadIdx.x;
    if (t < N * 64) out[t] = bias[t & 63];
    if (t < N * 4) { denom[t] = 0.f; emax_u[t] = 0u; }
}

__device__ __forceinline__ float leaky02(float x) { return x > 0.f ? x : 0.2f * x; }

// Monotone float -> uint map (unsigned compare == float compare).
__device__ __forceinline__ unsigned f2o(float f) {
    unsigned u = __float_as_uint(f);
    return (u >> 31) ? ~u : (u | 0x80000000u);
}

// ---------------------------------------------------------------------------
// Pass C: segment max over incoming edges (atomicMax on ordered uints).
// ---------------------------------------------------------------------------
__global__ void gat_emax(const int* __restrict__ src, const int* __restrict__ dst,
                         const float* __restrict__ el, const float* __restrict__ er,
                         unsigned* __restrict__ emax_u, int E)
{
    int t = blockIdx.x * blockDim.x + threadIdx.x;
    if (t >= E * 4) return;
    const int e = t >> 2, h = t & 3;
    const int s = src[e], d = dst[e];
    const float x = leaky02(el[s * 4 + h] + er[d * 4 + h]);
    atomicMax(&emax_u[d * 4 + h], f2o(x));
}

// Decode ordered uint -> float in place; untouched sentinel (zero in-degree) -> 0.
__global__ void gat_decode(unsigned* __restrict__ emax_u, float* __restrict__ emax_f, int N)
{
    int t = blockIdx.x * blockDim.x + threadIdx.x;
    if (t >= N * 4) return;
    const unsigned u = emax_u[t];
    float f;
    if (u == 0u)                 f = 0.f;
    else if (u & 0x80000000u)    f = __uint_as_float(u ^ 0x80000000u);
    else                         f = __uint_as_float(~u);
    emax_f[t] = f;
}

// ---------------------------------------------------------------------------
// Pass D: softmax denominator.
// ---------------------------------------------------------------------------
__global__ void gat_denom(const int* __restrict__ src, const int* __restrict__ dst,
                          const float* __restrict__ el, const float* __restrict__ er,
                          const float* __restrict__ emax, float* __restrict__ denom, int E)
{
    int t = blockIdx.x * blockDim.x + threadIdx.x;
    if (t >= E * 4) return;
    const int e = t >> 2, h = t & 3;
    const int s = src[e], d = dst[e];
    const int dh = d * 4 + h;
    const float x = leaky02(el[s * 4 + h] + er[dh]);
    atomicAdd(&denom[dh], __expf(x - emax[dh]));
}

// ---------------------------------------------------------------------------
// Pass E: weighted scatter-aggregate. 64 lanes per edge: lane c handles output
// channel c (head = c/16). Coalesced feat gather + coalesced atomicAdd.
// ---------------------------------------------------------------------------
__global__ void gat_agg(const int* __restrict__ src, const int* __restrict__ dst,
                        const float* __restrict__ el, const float* __restrict__ er,
                        const float* __restrict__ emax, const float* __restrict__ denom,
                        const float* __restrict__ feat, float* __restrict__ out, int E)
{
    int t = blockIdx.x * blockDim.x + threadIdx.x;
    if (t >= E * 64) return;
    const int e = t >> 6, c = t & 63, h = c >> 4;
    const int s = src[e], d = dst[e];
    const int dh = d * 4 + h;
    const float x = leaky02(el[s * 4 + h] + er[dh]);
    const float alpha = __expf(x - emax[dh]) / denom[dh];
    atomicAdd(&out[(size_t)d * 64 + c], feat[(size_t)s * 64 + c] * alpha);
}

// ---------------------------------------------------------------------------
extern "C" void kernel_launch(void* const* d_in, const int* in_sizes, int n_in,
                              void* d_out, int out_size, void* d_ws, size_t ws_size,
                              hipStream_t stream)
{
    const float* nf   = (const float*)d_in[0];
    const float* W    = (const float*)d_in[1];
    const float* al   = (const float*)d_in[2];
    const float* ar   = (const float*)d_in[3];
    const float* bias = (const float*)d_in[4];
    const int*   src  = (const int*)d_in[5];
    const int*   dst  = (const int*)d_in[6];

    const int N = in_sizes[0] / 64;
    const int E = in_sizes[5];

    float* ws    = (float*)d_ws;
    float* feat  = ws;                      // N*64
    float* el    = feat  + (size_t)N * 64;  // N*4
    float* er    = el    + (size_t)N * 4;   // N*4
    float* emax  = er    + (size_t)N * 4;   // N*4 (uint during max phase)
    float* denom = emax  + (size_t)N * 4;   // N*4
    float* out   = (float*)d_out;

    gat_proj_wmma<<<(N + 127) / 128, 256, 0, stream>>>(nf, W, feat, N);
    gat_elr<<<(N * 4 + 255) / 256, 256, 0, stream>>>(feat, al, ar, el, er, N);
    gat_init<<<(N * 64 + 255) / 256, 256, 0, stream>>>(out, bias, denom, (unsigned*)emax, N);
    gat_emax<<<(E * 4 + 255) / 256, 256, 0, stream>>>(src, dst, el, er, (unsigned*)emax, E);
    gat_decode<<<(N * 4 + 255) / 256, 256, 0, stream>>>((unsigned*)emax, emax, N);
    gat_denom<<<(E * 4 + 255) / 256, 256, 0, stream>>>(src, dst, el, er, emax, denom, E);
    {
        long long tot = (long long)E * 64;
        int blocks = (int)((tot + 255) / 256);
        gat_agg<<<blocks, 256, 0, stream>>>(src, dst, el, er, emax, denom, feat, out, E);
    }
}